// hCSSM_44332652430048
// MI455X (gfx1250) — compile-verified
//
#include <hip/hip_runtime.h>
#include <math.h>

typedef float v2f __attribute__((ext_vector_type(2)));
typedef float v8f __attribute__((ext_vector_type(8)));

#define WMMA_F32(a, b, c) \
  __builtin_amdgcn_wmma_f32_16x16x4_f32(false, (a), false, (b), (short)0, (c), false, false)

// ---------------- problem dims ----------------
constexpr int Bb = 2, Tt = 32, Hh = 65, Ww = 65, Cc = 32, Ff = 32, KSZ = 15;
constexpr int Wr = 33;        // W//2+1
constexpr int Hp = 33, Wp = 33;
constexpr int Wout = 64;      // 2*(Wr-1)
constexpr float TWO_PI = 6.283185307179586f;

// ---------------- workspace layout (floats) ----------------
constexpr size_t SZ_8068 = 80 * 68;
constexpr size_t SZ_6436 = 64 * 36;
constexpr size_t SZ_SPEC = (size_t)Hh * Wr * Cc;               // 68640
constexpr size_t SZ_GATE = (size_t)Bb * Wp * Cc;               // 2112
constexpr size_t SZ_U    = (size_t)Bb * Tt * Hh * 2 * Wr * Cc; // 8,785,920

constexpr size_t OFF_MWF   = 0;
constexpr size_t OFF_COSH  = OFF_MWF   + SZ_8068;
constexpr size_t OFF_SINH  = OFF_COSH  + SZ_8068;
constexpr size_t OFF_COSHI = OFF_SINH  + SZ_8068;
constexpr size_t OFF_SINHI = OFF_COSHI + SZ_8068;
constexpr size_t OFF_CWI   = OFF_SINHI + SZ_8068;
constexpr size_t OFF_SWI   = OFF_CWI   + SZ_6436;
constexpr size_t OFF_KERE  = OFF_SWI   + SZ_6436;
constexpr size_t OFF_KEIM  = OFF_KERE  + SZ_SPEC;
constexpr size_t OFF_KIRE  = OFF_KEIM  + SZ_SPEC;
constexpr size_t OFF_KIIM  = OFF_KIRE  + SZ_SPEC;
constexpr size_t OFF_CTX   = OFF_KIIM  + SZ_SPEC;
constexpr size_t OFF_ALPHA = OFF_CTX   + SZ_GATE;
constexpr size_t OFF_DELTA = OFF_ALPHA + SZ_GATE;
constexpr size_t OFF_MU    = OFF_DELTA + SZ_GATE;
constexpr size_t OFF_GAMMA = OFF_MU    + SZ_GATE;
constexpr size_t OFF_U1    = OFF_GAMMA + SZ_GATE;   // [(bt*65+h)][reim][wf][c], reused as Y1
constexpr size_t OFF_U2    = OFF_U1    + SZ_U;      // [bt][reim][hf][wf][c], Y written in place

// ---------------- WMMA fragment helpers ----------------
// 32-bit A 16x4 (ISA 7.12.2): lanes0-15 hold K=k0,k0+1; lanes16-31 hold K=k0+2,k0+3
__device__ __forceinline__ v2f ldA(const float* M, int stride, int rowBase,
                                   int k0, int lane) {
  int row = rowBase + (lane & 15);
  int ks  = k0 + ((lane >> 4) << 1);
  v2f a;
  a.x = M[row * stride + ks];
  a.y = M[row * stride + ks + 1];
  return a;
}

// B 4x16: rows striped across lanes within a VGPR (mirror of A's K split).
// Panels are zero-padded to the full K extent -> branchless LDS loads.
__device__ __forceinline__ v2f ldB(const float* X, int kstride, int k0, int lane) {
  int cc = lane & 15;
  int r0 = k0 + ((lane >> 4) << 1);
  v2f b;
  b.x = X[r0       * kstride + cc];
  b.y = X[(r0 + 1) * kstride + cc];
  return b;
}

// ---------------- twiddle / init kernel ----------------
__global__ __launch_bounds__(256) void init_tw(float* __restrict__ ws) {
  int idx = blockIdx.x * 256 + threadIdx.x;
  if (idx < (int)SZ_8068) {
    int row = idx / 68, col = idx % 68;
    float mw = 0.f, ch = 0.f, sh = 0.f, chi = 0.f, shi = 0.f;
    if (col < 65) {
      if (row < 33) {
        int p = (row * col) % 65;
        mw = cosf(TWO_PI * (float)p / 65.f);
      } else if (row < 66) {
        int p = ((row - 33) * col) % 65;
        mw = -sinf(TWO_PI * (float)p / 65.f);
      }
      if (row < 65) {
        int p = (row * col) % 65;
        float th = TWO_PI * (float)p / 65.f;
        ch = cosf(th); sh = sinf(th);
        chi = ch * (1.f / 65.f); shi = sh * (1.f / 65.f);
      }
    }
    ws[OFF_MWF + idx] = mw;
    ws[OFF_COSH + idx] = ch;  ws[OFF_SINH + idx] = sh;
    ws[OFF_COSHI + idx] = chi; ws[OFF_SINHI + idx] = shi;
  }
  if (idx < (int)SZ_6436) {
    int row = idx / 36, k = idx % 36;
    float cw = 0.f, sw = 0.f;
    if (k == 0) cw = 1.f / 64.f;
    else if (k == 32) cw = ((row & 1) ? -1.f : 1.f) / 64.f;
    else if (k < 32) {
      int p = (row * k) % 64;
      float th = TWO_PI * (float)p / 64.f;
      cw = 2.f * cosf(th) / 64.f;
      sw = -2.f * sinf(th) / 64.f;
    }
    ws[OFF_CWI + idx] = cw;
    ws[OFF_SWI + idx] = sw;
  }
  if (idx < (int)SZ_GATE) ws[OFF_CTX + idx] = 0.f;
}

// ---------------- conv3x3 s2 + gelu + ctx-sum ----------------
__global__ __launch_bounds__(256) void conv_ctx(const float* __restrict__ in,
                                                const float* __restrict__ w,
                                                const float* __restrict__ bias,
                                                float* __restrict__ ctx) {
  int idx = blockIdx.x * 256 + threadIdx.x;
  if (idx >= Bb * Tt * Hp * Wp * Ff) return;
  int f  = idx & 31;
  int wp = (idx >> 5) % Wp;
  int hp = (idx / (Ff * Wp)) % Hp;
  int bt = idx / (Ff * Wp * Hp);
  float acc = bias[f];
  for (int kh = 0; kh < 3; kh++) {
    int hi = hp * 2 + kh - 1;
    if ((unsigned)hi >= (unsigned)Hh) continue;
    for (int kw = 0; kw < 3; kw++) {
      int wi = wp * 2 + kw - 1;
      if ((unsigned)wi >= (unsigned)Ww) continue;
      const float* ip = in + (((size_t)bt * Hh + hi) * Ww + wi) * Cc;
      const float* wq = w + ((size_t)(kh * 3 + kw) * Cc) * Ff + f;
#pragma unroll
      for (int ci = 0; ci < Cc; ci++) acc += ip[ci] * wq[(size_t)ci * Ff];
    }
  }
  float x = acc;
  float g = 0.5f * x * (1.f + tanhf(0.7978845608028654f * (x + 0.044715f * x * x * x)));
  int b = bt / Tt;
  atomicAdd(&ctx[(b * Wp + wp) * Ff + f], g);
}

// ---------------- gate matmuls + activations ----------------
__global__ __launch_bounds__(256) void gates_k(const float* __restrict__ ctx,
    const float* __restrict__ wA, const float* __restrict__ bA,
    const float* __restrict__ wD, const float* __restrict__ bD,
    const float* __restrict__ wM, const float* __restrict__ bM,
    const float* __restrict__ wG, const float* __restrict__ bG,
    float* __restrict__ alpha, float* __restrict__ delta,
    float* __restrict__ mu, float* __restrict__ gamma) {
  int idx = blockIdx.x * 256 + threadIdx.x;
  if (idx >= (int)SZ_GATE) return;
  int c  = idx & 31;
  int wp = (idx >> 5) % Wp;
  int b  = idx / (Cc * Wp);
  const float* cv = ctx + (b * Wp + wp) * Ff;
  const float scale = 1.f / (float)(Tt * Hp);
  float aA = bA[c], aD = bD[c], aM = bM[c], aG = bG[c];
#pragma unroll
  for (int f = 0; f < Ff; f++) {
    float v = cv[f] * scale;
    aA += v * wA[f * Cc + c];
    aD += v * wD[f * Cc + c];
    aM += v * wM[f * Cc + c];
    aG += v * wG[f * Cc + c];
  }
  alpha[idx] = 1.f / (1.f + expf(-aA));
  delta[idx] = 1.f / (1.f + expf(-aD));
  mu[idx]    = (aM > 30.f) ? aM : log1pf(expf(aM));
  gamma[idx] = (aG > 30.f) ? aG : log1pf(expf(aG));
}

// ---------------- kernel spectra (direct 225-tap DFT) ----------------
__global__ __launch_bounds__(256) void spectra_k(const float* __restrict__ kE,
                                                 const float* __restrict__ kI,
                                                 float* __restrict__ KEre, float* __restrict__ KEim,
                                                 float* __restrict__ KIre, float* __restrict__ KIim) {
  int idx = blockIdx.x * 256 + threadIdx.x;
  if (idx >= (int)SZ_SPEC) return;
  int c  = idx & 31;
  int wf = (idx >> 5) % Wr;
  int hf = idx / (Cc * Wr);
  float er = 0.f, ei = 0.f, ir = 0.f, ii = 0.f;
  for (int kh = 0; kh < KSZ; kh++) {
    for (int kw = 0; kw < KSZ; kw++) {
      int p = (hf * (kh + 25) + wf * (kw + 25)) % 65;
      float th = TWO_PI * (float)p / 65.f;
      float co = cosf(th), sn = sinf(th);
      float ke = kE[(kh * KSZ + kw) * Cc + c];
      float ki = kI[(kh * KSZ + kw) * Cc + c];
      er += ke * co; ei -= ke * sn;
      ir += ki * co; ii -= ki * sn;
    }
  }
  KEre[idx] = er; KEim[idx] = ei;
  KIre[idx] = ir; KIim[idx] = ii;
}

// ---------------- forward W-DFT (real -> complex), LDS-staged WMMA GEMM ------
// one block per slice (bt*65+h); 8 waves cover 10 tiles (5 mt x 2 nt)
__global__ __launch_bounds__(256) void dftW_fwd(const float* __restrict__ in,
                                                const float* __restrict__ MwF,
                                                float* __restrict__ U1) {
  __shared__ float sA[80 * 68];    // combined cos/-sin twiddle, zero-padded
  __shared__ float sX[68 * Cc];    // input panel, rows 65..67 zero
  int tid = threadIdx.x;
  int slice = blockIdx.x;          // 0..4159
  const float* X = in + (size_t)slice * Ww * Cc;
  for (int i = tid; i < 80 * 68; i += 256) sA[i] = MwF[i];
  for (int i = tid; i < 68 * Cc; i += 256) sX[i] = (i < Ww * Cc) ? X[i] : 0.f;
  __syncthreads();

  int wv = tid >> 5, lane = tid & 31;
  float* base = U1 + (size_t)slice * 2 * Wr * Cc;
  for (int tile = wv; tile < 10; tile += 8) {
    int mt = tile >> 1, nt = tile & 1;
    v8f acc = {};
    for (int k0 = 0; k0 < 68; k0 += 4) {
      v2f a = ldA(sA, 68, 16 * mt, k0, lane);
      v2f b = ldB(sX + nt * 16, Cc, k0, lane);
      acc = WMMA_F32(a, b, acc);
    }
    int half = lane >> 4, col = nt * 16 + (lane & 15);
#pragma unroll
    for (int r = 0; r < 8; r++) {
      int row = 16 * mt + half * 8 + r;
      float v = acc[r];
      if (row < 33)      base[(size_t)row * Cc + col] = v;
      else if (row < 66) base[(size_t)Wr * Cc + (size_t)(row - 33) * Cc + col] = v;
    }
  }
}

// ---------------- forward H-DFT (complex -> complex), LDS-staged -------------
// slice = bt*Wr + wf; re = cos*re + sin*im; im = cos*im - sin*re   (e^{-i})
__global__ __launch_bounds__(256) void dftH_fwd(const float* __restrict__ U1,
                                                const float* __restrict__ CosH,
                                                const float* __restrict__ SinH,
                                                float* __restrict__ U2) {
  __shared__ float sC[80 * 68];
  __shared__ float sS[80 * 68];
  __shared__ float sP[68 * 64];    // panel: [h][0:32]=re, [h][32:64]=im; rows 65..67 zero
  int tid = threadIdx.x;
  int slice = blockIdx.x;          // 0..2111
  int bt = slice / Wr, wf = slice - bt * Wr;
  for (int i = tid; i < 80 * 68; i += 256) { sC[i] = CosH[i]; sS[i] = SinH[i]; }
  const float* src = U1 + (size_t)bt * Hh * 2 * Wr * Cc + (size_t)wf * Cc;
  for (int i = tid; i < 68 * 64; i += 256) {
    int h = i >> 6, q = i & 63;                 // q<32: re, q>=32: im
    int reim = q >> 5, c = q & 31;
    sP[i] = (h < Hh) ? src[(size_t)h * 2 * Wr * Cc + (size_t)reim * Wr * Cc + c] : 0.f;
  }
  __syncthreads();

  int wv = tid >> 5, lane = tid & 31;
  float* oRe = U2 + (size_t)bt * 2 * Hh * Wr * Cc + (size_t)wf * Cc;
  float* oIm = oRe + (size_t)Hh * Wr * Cc;
  for (int tile = wv; tile < 10; tile += 8) {
    int mt = tile >> 1, nt = tile & 1;
    v8f cre = {}, cim = {};
    for (int k0 = 0; k0 < 68; k0 += 4) {
      v2f aC = ldA(sC, 68, 16 * mt, k0, lane);
      v2f aS = ldA(sS, 68, 16 * mt, k0, lane);
      v2f bR = ldB(sP + nt * 16,      64, k0, lane);
      v2f bI = ldB(sP + 32 + nt * 16, 64, k0, lane);
      cre = WMMA_F32(aC, bR, cre);
      cre = WMMA_F32(aS, bI, cre);
      cim = WMMA_F32(aC, bI, cim);
      v2f aSn = -aS;
      cim = WMMA_F32(aSn, bR, cim);
    }
    int half = lane >> 4, col = nt * 16 + (lane & 15);
#pragma unroll
    for (int r = 0; r < 8; r++) {
      int hf = 16 * mt + half * 8 + r;
      if (hf < Hh) {
        oRe[(size_t)hf * Wr * Cc + col] = cre[r];
        oIm[(size_t)hf * Wr * Cc + col] = cim[r];
      }
    }
  }
}

// ---------------- time-invariant SSM scan (Y written in place over U2) -------
__global__ __launch_bounds__(256) void ssm_scan(const float* __restrict__ alpha,
    const float* __restrict__ delta, const float* __restrict__ mu,
    const float* __restrict__ gamma,
    const float* __restrict__ KEre, const float* __restrict__ KEim,
    const float* __restrict__ KIre, const float* __restrict__ KIim,
    float* __restrict__ U2) {
  int idx = blockIdx.x * 256 + threadIdx.x;
  if (idx >= Bb * Hh * Wr * Cc) return;
  int c  = idx & 31;
  int wf = (idx >> 5) % Wr;
  int hf = (idx / (Cc * Wr)) % Hh;
  int b  = idx / (Cc * Wr * Hh);
  int gidx = (b * Wr + wf) * Cc + c;
  int kidx = (hf * Wr + wf) * Cc + c;
  float axx = 0.9f * alpha[gidx], ayy = 0.9f * delta[gidx];
  float m = mu[gidx], g = gamma[gidx];
  float axy_re = -KIre[kidx] * m, axy_im = -KIim[kidx] * m;
  float ayx_re =  KEre[kidx] * g, ayx_im =  KEim[kidx] * g;
  size_t planeBT = (size_t)2 * Hh * Wr * Cc;
  size_t imOff   = (size_t)Hh * Wr * Cc;
  size_t base = (size_t)(b * Tt) * planeBT + (size_t)hf * Wr * Cc + (size_t)wf * Cc + c;
  float sxr = 0.f, sxi = 0.f, syr = 0.f, syi = 0.f;
  for (int t = 0; t < Tt; t++) {
    size_t p = base + (size_t)t * planeBT;
    float ur = U2[p], ui = U2[p + imOff];
    float nxr = axx * sxr + axy_re * syr - axy_im * syi + ur;
    float nxi = axx * sxi + axy_re * syi + axy_im * syr + ui;
    float nyr = ayx_re * sxr - ayx_im * sxi + ayy * syr;
    float nyi = ayx_re * sxi + ayx_im * sxr + ayy * syi;
    U2[p] = nyr; U2[p + imOff] = nyi;
    sxr = nxr; sxi = nxi; syr = nyr; syi = nyi;
  }
}

// ---------------- inverse H-DFT (complex, 1/65 folded), LDS-staged -----------
// re = cos*re - sin*im; im = cos*im + sin*re   (e^{+i})
__global__ __launch_bounds__(256) void idftH(const float* __restrict__ U2,
                                             const float* __restrict__ CosHI,
                                             const float* __restrict__ SinHI,
                                             float* __restrict__ Y1) {
  __shared__ float sC[80 * 68];
  __shared__ float sS[80 * 68];
  __shared__ float sP[68 * 64];
  int tid = threadIdx.x;
  int slice = blockIdx.x;          // 0..2111
  int bt = slice / Wr, wf = slice - bt * Wr;
  for (int i = tid; i < 80 * 68; i += 256) { sC[i] = CosHI[i]; sS[i] = SinHI[i]; }
  const float* src = U2 + (size_t)bt * 2 * Hh * Wr * Cc + (size_t)wf * Cc;
  for (int i = tid; i < 68 * 64; i += 256) {
    int h = i >> 6, q = i & 63;
    int reim = q >> 5, c = q & 31;
    sP[i] = (h < Hh) ? src[(size_t)reim * Hh * Wr * Cc + (size_t)h * Wr * Cc + c] : 0.f;
  }
  __syncthreads();

  int wv = tid >> 5, lane = tid & 31;
  for (int tile = wv; tile < 10; tile += 8) {
    int mt = tile >> 1, nt = tile & 1;
    v8f cre = {}, cim = {};
    for (int k0 = 0; k0 < 68; k0 += 4) {
      v2f aC = ldA(sC, 68, 16 * mt, k0, lane);
      v2f aS = ldA(sS, 68, 16 * mt, k0, lane);
      v2f bR = ldB(sP + nt * 16,      64, k0, lane);
      v2f bI = ldB(sP + 32 + nt * 16, 64, k0, lane);
      cre = WMMA_F32(aC, bR, cre);
      v2f aSn = -aS;
      cre = WMMA_F32(aSn, bI, cre);
      cim = WMMA_F32(aC, bI, cim);
      cim = WMMA_F32(aS, bR, cim);
    }
    int half = lane >> 4, col = nt * 16 + (lane & 15);
#pragma unroll
    for (int r = 0; r < 8; r++) {
      int h = 16 * mt + half * 8 + r;
      if (h < Hh) {
        float* base = Y1 + ((size_t)(bt * Hh + h) * 2) * Wr * Cc + (size_t)wf * Cc;
        base[col] = cre[r];
        base[(size_t)Wr * Cc + col] = cim[r];
      }
    }
  }
}

// ---------------- irfft over wf (complex 33 -> real 64), LDS-staged ----------
__global__ __launch_bounds__(256) void idftW(const float* __restrict__ Y1,
                                             const float* __restrict__ CwI,
                                             const float* __restrict__ SwI,
                                             float* __restrict__ out) {
  __shared__ float sC[64 * 36];
  __shared__ float sS[64 * 36];
  __shared__ float sP[2 * 36 * Cc];    // [reim][wf(pad36)][c], rows 33..35 zero
  int tid = threadIdx.x;
  int slice = blockIdx.x;              // bt*65 + h, 0..4159
  const float* src = Y1 + (size_t)slice * 2 * Wr * Cc;
  for (int i = tid; i < 64 * 36; i += 256) { sC[i] = CwI[i]; sS[i] = SwI[i]; }
  for (int i = tid; i < 2 * 36 * Cc; i += 256) {
    int reim = i / (36 * Cc);
    int rem  = i - reim * (36 * Cc);
    int wfr  = rem >> 5, c = rem & 31;
    sP[i] = (wfr < Wr) ? src[(size_t)reim * Wr * Cc + (size_t)wfr * Cc + c] : 0.f;
  }
  __syncthreads();

  int wv = tid >> 5, lane = tid & 31;  // 8 waves, exactly 8 tiles (4 mt x 2 nt)
  int mt = wv >> 1, nt = wv & 1;
  v8f acc = {};
  for (int k0 = 0; k0 < 36; k0 += 4) {
    v2f aC = ldA(sC, 36, 16 * mt, k0, lane);
    v2f aS = ldA(sS, 36, 16 * mt, k0, lane);
    v2f bR = ldB(sP + nt * 16,            Cc, k0, lane);
    v2f bI = ldB(sP + 36 * Cc + nt * 16,  Cc, k0, lane);
    acc = WMMA_F32(aC, bR, acc);
    acc = WMMA_F32(aS, bI, acc);
  }
  int half = lane >> 4, col = nt * 16 + (lane & 15);
  float* ob = out + (size_t)slice * Wout * Cc;
#pragma unroll
  for (int r = 0; r < 8; r++) {
    int w = 16 * mt + half * 8 + r;    // always < 64
    ob[(size_t)w * Cc + col] = acc[r];
  }
}

// ---------------- host launcher ----------------
extern "C" void kernel_launch(void* const* d_in, const int* in_sizes, int n_in,
                              void* d_out, int out_size, void* d_ws, size_t ws_size,
                              hipStream_t stream) {
  const float* in_seq = (const float*)d_in[0];
  const float* conv_w = (const float*)d_in[1];
  const float* conv_b = (const float*)d_in[2];
  const float* w_alpha = (const float*)d_in[3];
  const float* b_alpha = (const float*)d_in[4];
  const float* w_delta = (const float*)d_in[5];
  const float* b_delta = (const float*)d_in[6];
  const float* w_mu    = (const float*)d_in[7];
  const float* b_mu    = (const float*)d_in[8];
  const float* w_gamma = (const float*)d_in[9];
  const float* b_gamma = (const float*)d_in[10];
  const float* k_exc   = (const float*)d_in[11];
  const float* k_inh   = (const float*)d_in[12];

  float* ws = (float*)d_ws;
  float* out = (float*)d_out;

  float* MwF   = ws + OFF_MWF;
  float* CosH  = ws + OFF_COSH;
  float* SinH  = ws + OFF_SINH;
  float* CosHI = ws + OFF_COSHI;
  float* SinHI = ws + OFF_SINHI;
  float* CwI   = ws + OFF_CWI;
  float* SwI   = ws + OFF_SWI;
  float* KEre  = ws + OFF_KERE;
  float* KEim  = ws + OFF_KEIM;
  float* KIre  = ws + OFF_KIRE;
  float* KIim  = ws + OFF_KIIM;
  float* Ctx   = ws + OFF_CTX;
  float* Alpha = ws + OFF_ALPHA;
  float* Delta = ws + OFF_DELTA;
  float* Mu    = ws + OFF_MU;
  float* Gamma = ws + OFF_GAMMA;
  float* U1    = ws + OFF_U1;
  float* U2    = ws + OFF_U2;

  // 1) twiddles + zero ctx accumulator
  init_tw<<<(int)((SZ_8068 + 255) / 256), 256, 0, stream>>>(ws);

  // 2) conv + gelu + ctx-sum (atomics), then gates
  conv_ctx<<<(Bb * Tt * Hp * Wp * Ff + 255) / 256, 256, 0, stream>>>(in_seq, conv_w, conv_b, Ctx);
  gates_k<<<((int)SZ_GATE + 255) / 256, 256, 0, stream>>>(Ctx, w_alpha, b_alpha, w_delta, b_delta,
                                                          w_mu, b_mu, w_gamma, b_gamma,
                                                          Alpha, Delta, Mu, Gamma);

  // 3) kernel spectra
  spectra_k<<<((int)SZ_SPEC + 255) / 256, 256, 0, stream>>>(k_exc, k_inh, KEre, KEim, KIre, KIim);

  // 4) forward rfft2 via LDS-staged WMMA GEMMs (one block per slice)
  dftW_fwd<<<Bb * Tt * Hh, 256, 0, stream>>>(in_seq, MwF, U1);        // 4160 blocks
  dftH_fwd<<<Bb * Tt * Wr, 256, 0, stream>>>(U1, CosH, SinH, U2);     // 2112 blocks

  // 5) SSM recurrence over T (in place in U2)
  ssm_scan<<<(Bb * Hh * Wr * Cc + 255) / 256, 256, 0, stream>>>(Alpha, Delta, Mu, Gamma,
                                                                KEre, KEim, KIre, KIim, U2);

  // 6) inverse: H pass (into U1 space reused as Y1), then irfft over wf to d_out
  idftH<<<Bb * Tt * Wr, 256, 0, stream>>>(U2, CosHI, SinHI, U1);      // 2112 blocks
  idftW<<<Bb * Tt * Hh, 256, 0, stream>>>(U1, CwI, SwI, out);         // 4160 blocks
}